// gcn_operation_45561013076437
// MI455X (gfx1250) — compile-verified
//
#include <hip/hip_runtime.h>
#include <hip/hip_bf16.h>
#include <cmath>

typedef float v2f __attribute__((ext_vector_type(2)));
typedef float v8f __attribute__((ext_vector_type(8)));
typedef int   vi4 __attribute__((vector_size(4 * sizeof(int))));

#define NN    2048
#define TT    12
#define BBATCH 16
#define CINC  64
#define COUTC 64
#define NCOL  1024          // BBATCH * CINC

// ------------------------------------------------------------------
// Kernel 1: d_static[n] = 1/sqrt(1 + sum_m relu(A_static[n,m]))
// ------------------------------------------------------------------
__global__ __launch_bounds__(256) void k_deg_static(const float* __restrict__ As,
                                                    float* __restrict__ ds) {
    __shared__ float red[256];
    const int n = blockIdx.x;
    float s = 0.f;
    for (int m = threadIdx.x; m < NN; m += 256)
        s += fmaxf(As[(size_t)n * NN + m], 0.f);
    red[threadIdx.x] = s;
    __syncthreads();
    for (int off = 128; off > 0; off >>= 1) {
        if (threadIdx.x < off) red[threadIdx.x] += red[threadIdx.x + off];
        __syncthreads();
    }
    if (threadIdx.x == 0) ds[n] = 1.0f / sqrtf(red[0] + 1.0f);
}

// ------------------------------------------------------------------
// Kernel 2: build combined normalized adjacency row-by-row.
//  adaptive row degree = 1 (softmax) + 1 (diag) = 2 -> d_a*d_a = 0.5
// ------------------------------------------------------------------
__global__ __launch_bounds__(256) void k_build_A(const float* __restrict__ As,
                                                 const float* __restrict__ e1g,
                                                 const float* __restrict__ e2g,
                                                 const float* __restrict__ alpha,
                                                 const float* __restrict__ ds,
                                                 float* __restrict__ Aout) {
    __shared__ float sl[NN];
    __shared__ float red[256];
    __shared__ float e1[16];
    const int n = blockIdx.x;
    const int tid = threadIdx.x;
    if (tid < 16) e1[tid] = e1g[n * 16 + tid];
    __syncthreads();

    float lmax = 0.f;   // logits are relu'd -> >= 0
    for (int m = tid; m < NN; m += 256) {
        float d = 0.f;
        #pragma unroll
        for (int k = 0; k < 16; ++k) d += e1[k] * e2g[k * NN + m];
        d = fmaxf(d, 0.f);
        sl[m] = d;
        lmax = fmaxf(lmax, d);
    }
    red[tid] = lmax; __syncthreads();
    for (int off = 128; off > 0; off >>= 1) {
        if (tid < off) red[tid] = fmaxf(red[tid], red[tid + off]);
        __syncthreads();
    }
    const float mx = red[0];
    __syncthreads();

    float ps = 0.f;
    for (int m = tid; m < NN; m += 256) ps += expf(sl[m] - mx);
    red[tid] = ps; __syncthreads();
    for (int off = 128; off > 0; off >>= 1) {
        if (tid < off) red[tid] += red[tid + off];
        __syncthreads();
    }
    const float inv_s = 1.0f / red[0];

    const float a  = 1.0f / (1.0f + expf(-alpha[0]));
    const float dn = ds[n];
    for (int m = tid; m < NN; m += 256) {
        const float diag = (m == n) ? 1.f : 0.f;
        const float stat = fmaxf(As[(size_t)n * NN + m], 0.f) + diag;
        const float adap = expf(sl[m] - mx) * inv_s + diag;
        Aout[(size_t)n * NN + m] = a * dn * ds[m] * stat + (1.f - a) * 0.5f * adap;
    }
}

// ------------------------------------------------------------------
// Kernel 3: fused  Y_t = A @ X_t  ->  FC (64->128) -> bias -> GLU
//  block: 256 threads = 8 wave32s; tile 64(n) x 128(b*c); K over m.
//  Double-buffered ASYNC global->LDS staging (ASYNCcnt), WMMA f32.
// ------------------------------------------------------------------
#define KC    32
#define SA_LD 36     // 16B-aligned row stride, spreads banks
#define SX_LD 132
#define SY_LD 132
#define STAGE_FLTS (64 * SA_LD + KC * SX_LD)   // 6528 floats per buffer

__device__ __forceinline__ void async_cp16(const float* gsrc, float* ldst) {
    __builtin_amdgcn_global_load_async_to_lds_b128(
        (__attribute__((address_space(1))) vi4*)(gsrc),
        (__attribute__((address_space(3))) vi4*)(ldst), 0, 0);
}

__global__ __launch_bounds__(256) void k_gcn_main(const float* __restrict__ A,
                                                  const float* __restrict__ x,
                                                  const float* __restrict__ W,
                                                  const float* __restrict__ bias,
                                                  float* __restrict__ out) {
    // two stage buffers (ping-pong), Y tile aliases them after the main loop
    __shared__ __align__(16) float smem[2 * STAGE_FLTS];   // 52224 B
    float* sY = smem;                                      // [64][SY_LD] (33792 B)

    const int t    = blockIdx.z;
    const int n0   = blockIdx.y * 64;
    const int col0 = blockIdx.x * 128;
    const int b0   = col0 >> 6;          // 2 batch slots per block
    const int tid  = threadIdx.x;
    const int lane = tid & 31;
    const int wid  = tid >> 5;
    const int lrow = lane & 15;          // N / M index inside 16
    const int lhi  = lane >> 4;          // lane-half

    const int wm = wid >> 2;             // 0..1 row group
    const int wn = wid & 3;              // 0..3 col group

    const float* xt = x + (size_t)t * NN * NCOL;

    // per-thread async copy slots (2 for A tile, 4 for X tile = 6/chunk)
    const int ar0 = tid >> 3, ac0 = (tid & 7) << 2;              // A slot 0
    const int ar1 = (tid + 256) >> 3, ac1 = ac0;                 // A slot 1
    const int xr0 = tid >> 5, xc0 = (tid & 31) << 2;             // X slots 0..3 (+8 rows each)

    auto issue_chunk = [&](float* buf, int k0) {
        float* bA = buf;                  // [64][SA_LD]
        float* bX = buf + 64 * SA_LD;     // [KC][SX_LD]
        async_cp16(&A[(size_t)(n0 + ar0) * NN + k0 + ac0], &bA[ar0 * SA_LD + ac0]);
        async_cp16(&A[(size_t)(n0 + ar1) * NN + k0 + ac1], &bA[ar1 * SA_LD + ac1]);
        #pragma unroll
        for (int it = 0; it < 4; ++it) {
            const int r = xr0 + it * 8;
            async_cp16(&xt[(size_t)(k0 + r) * NCOL + col0 + xc0], &bX[r * SX_LD + xc0]);
        }
    };

    v8f acc[2][2];
    #pragma unroll
    for (int i = 0; i < 2; ++i)
        #pragma unroll
        for (int j = 0; j < 2; ++j)
            acc[i][j] = v8f{0.f,0.f,0.f,0.f,0.f,0.f,0.f,0.f};

    issue_chunk(smem, 0);                       // prologue: chunk 0 -> buffer 0

    const int NCHUNK = NN / KC;                 // 64
    for (int kc = 0; kc < NCHUNK; ++kc) {
        float* cur = smem + (kc & 1) * STAGE_FLTS;
        if (kc + 1 < NCHUNK) {
            issue_chunk(smem + ((kc + 1) & 1) * STAGE_FLTS, (kc + 1) * KC);
            __builtin_amdgcn_s_wait_asynccnt(6);   // current chunk landed; next in flight
        } else {
            __builtin_amdgcn_s_wait_asynccnt(0);
        }
        __syncthreads();

        float* sA = cur;
        float* sX = cur + 64 * SA_LD;
        #pragma unroll
        for (int kb = 0; kb < KC; kb += 4) {
            v2f af[2], bf[2];
            #pragma unroll
            for (int sm = 0; sm < 2; ++sm) {
                const int r = wm * 32 + sm * 16 + lrow;
                af[sm] = *(const v2f*)&sA[r * SA_LD + kb + 2 * lhi];   // K = v + 2*(lane/16)
            }
            #pragma unroll
            for (int sn = 0; sn < 2; ++sn) {
                const int c = wn * 32 + sn * 16 + lrow;
                bf[sn][0] = sX[(kb + 2 * lhi)     * SX_LD + c];
                bf[sn][1] = sX[(kb + 2 * lhi + 1) * SX_LD + c];
            }
            #pragma unroll
            for (int sm = 0; sm < 2; ++sm)
                #pragma unroll
                for (int sn = 0; sn < 2; ++sn)
                    acc[sm][sn] = __builtin_amdgcn_wmma_f32_16x16x4_f32(
                        false, af[sm], false, bf[sn], (short)0, acc[sm][sn], false, false);
        }
        __syncthreads();
    }

    // spill Y tile (64 x 128) to LDS; C layout: M = v + 8*lhi, N = lrow
    #pragma unroll
    for (int sm = 0; sm < 2; ++sm)
        #pragma unroll
        for (int sn = 0; sn < 2; ++sn)
            #pragma unroll
            for (int v = 0; v < 8; ++v) {
                const int r = wm * 32 + sm * 16 + v + 8 * lhi;
                const int c = wn * 32 + sn * 16 + lrow;
                sY[r * SY_LD + c] = acc[sm][sn][v];
            }
    __syncthreads();

    // ---- FC (rows = (n,b) pairs: 128, K = c: 64, cols j: 128) + GLU ----
    const int rb = (wid >> 1) * 32;      // 4 row groups
    const int jb = (wid & 1) * 32;       // 2 lhs-col groups; rhs = +64

    v8f aL[2][2], aR[2][2];
    #pragma unroll
    for (int i = 0; i < 2; ++i)
        #pragma unroll
        for (int j = 0; j < 2; ++j) {
            aL[i][j] = v8f{0.f,0.f,0.f,0.f,0.f,0.f,0.f,0.f};
            aR[i][j] = v8f{0.f,0.f,0.f,0.f,0.f,0.f,0.f,0.f};
        }

    #pragma unroll
    for (int cb = 0; cb < CINC; cb += 4) {
        const int c0 = cb + 2 * lhi;
        v2f yf[2], wl[2], wr[2];
        #pragma unroll
        for (int sm = 0; sm < 2; ++sm) {
            const int r  = rb + sm * 16 + lrow;
            const int nl = r >> 1, bl = r & 1;
            yf[sm] = *(const v2f*)&sY[nl * SY_LD + bl * 64 + c0];
        }
        #pragma unroll
        for (int sn = 0; sn < 2; ++sn) {
            const int j = jb + sn * 16 + lrow;
            wl[sn][0] = W[(size_t)c0       * 128 + j];
            wl[sn][1] = W[(size_t)(c0 + 1) * 128 + j];
            wr[sn][0] = W[(size_t)c0       * 128 + j + 64];
            wr[sn][1] = W[(size_t)(c0 + 1) * 128 + j + 64];
        }
        #pragma unroll
        for (int sm = 0; sm < 2; ++sm)
            #pragma unroll
            for (int sn = 0; sn < 2; ++sn) {
                aL[sm][sn] = __builtin_amdgcn_wmma_f32_16x16x4_f32(
                    false, yf[sm], false, wl[sn], (short)0, aL[sm][sn], false, false);
                aR[sm][sn] = __builtin_amdgcn_wmma_f32_16x16x4_f32(
                    false, yf[sm], false, wr[sn], (short)0, aR[sm][sn], false, false);
            }
    }

    // bias + GLU + store (lhs/rhs fragments share identical (row,col) mapping)
    #pragma unroll
    for (int sm = 0; sm < 2; ++sm)
        #pragma unroll
        for (int sn = 0; sn < 2; ++sn) {
            const int jl = jb + sn * 16 + lrow;
            const float bl_ = bias[jl];
            const float br_ = bias[jl + 64];
            #pragma unroll
            for (int v = 0; v < 8; ++v) {
                const int r = rb + sm * 16 + v + 8 * lhi;
                const int n = n0 + (r >> 1);
                const int b = b0 + (r & 1);
                const float l  = aL[sm][sn][v] + bl_;
                const float rr = aR[sm][sn][v] + br_;
                out[((size_t)(t * NN + n) * BBATCH + b) * COUTC + jl] =
                    l * (1.0f / (1.0f + expf(-rr)));
            }
        }
}

// ------------------------------------------------------------------
extern "C" void kernel_launch(void* const* d_in, const int* in_sizes, int n_in,
                              void* d_out, int out_size, void* d_ws, size_t ws_size,
                              hipStream_t stream) {
    const float* x   = (const float*)d_in[0];
    const float* As  = (const float*)d_in[1];
    const float* e1  = (const float*)d_in[2];
    const float* e2  = (const float*)d_in[3];
    const float* alp = (const float*)d_in[4];
    const float* W   = (const float*)d_in[5];
    const float* bfc = (const float*)d_in[6];
    float* out = (float*)d_out;

    float* wsA  = (float*)d_ws;                    // [NN*NN] combined adjacency (16 MB)
    float* wsDs = wsA + (size_t)NN * NN;           // [NN] static inv-sqrt degrees

    k_deg_static<<<NN, 256, 0, stream>>>(As, wsDs);
    k_build_A  <<<NN, 256, 0, stream>>>(As, e1, e2, alp, wsDs, wsA);

    dim3 grid(NCOL / 128, NN / 64, TT);
    k_gcn_main<<<grid, 256, 0, stream>>>(wsA, x, W, bfc, out);
}